// JATAdjacency_79783312491271
// MI455X (gfx1250) — compile-verified
//
#include <hip/hip_runtime.h>

typedef float v2f __attribute__((ext_vector_type(2)));
typedef float v8f __attribute__((ext_vector_type(8)));
typedef int v4i_vs __attribute__((vector_size(16)));  // matches builtin param

#define GLOBAL_AS __attribute__((address_space(1)))
#define LDS_AS    __attribute__((address_space(3)))

#define L_SEQ    1024
#define D_MODEL  1024
#define HEAD_DIM 64
#define NUM_HEADS 16
#define NUM_JAT   8
#define BATCH     4
#define LLSZ (1u << 20)  // L*L

#define BM 128
#define BN 128
#define BK 16

enum { MODE_PLAIN = 0, MODE_SCORES = 1, MODE_ATILDE = 2, MODE_PGEMM = 3 };

__device__ __forceinline__ v8f wmma4(v2f a, v2f b, v8f c) {
  // D = A(16x4,f32) * B(4x16,f32) + C(16x16,f32)
  return __builtin_amdgcn_wmma_f32_16x16x4_f32(false, a, false, b, (short)0, c,
                                               false, false);
}

__device__ __forceinline__ v2f lds2(const float* p) {
  return *reinterpret_cast<const v2f*>(p);  // 8B-aligned -> ds_load_b64
}

// ---- CDNA5 async global->LDS staging (ASYNCcnt path), with VGPR fallback ----
#if __has_builtin(__builtin_amdgcn_global_load_async_to_lds_b128)
#define HAVE_ASYNC_LDS 1
__device__ __forceinline__ void async_copy16(const float* g, float* l) {
  // param1: int4* (global source), param2: int4* (LDS dest), imm offset, cpol.
  // Pass AS1/AS3-qualified pointers: exact match if params are AS-qualified,
  // legal implicit subset->generic conversion otherwise.
  GLOBAL_AS v4i_vs* gp = (GLOBAL_AS v4i_vs*)(__UINTPTR_TYPE__)g;
  LDS_AS v4i_vs* lp = (LDS_AS v4i_vs*)(LDS_AS void*)l;
  __builtin_amdgcn_global_load_async_to_lds_b128(gp, lp, 0, 0);
}
__device__ __forceinline__ void async_wait0() {
#if __has_builtin(__builtin_amdgcn_s_wait_asynccnt)
  __builtin_amdgcn_s_wait_asynccnt(0);
#else
  asm volatile("s_wait_asynccnt 0x0" ::: "memory");
#endif
}
#else
#define HAVE_ASYNC_LDS 0
#endif

__device__ __forceinline__ void stage8(const float* g, float* l) {
#if HAVE_ASYNC_LDS
  async_copy16(g, l);
  async_copy16(g + 4, l + 4);
#else
  const float4 a = *reinterpret_cast<const float4*>(g);
  const float4 b = *reinterpret_cast<const float4*>(g + 4);
  *reinterpret_cast<float4*>(l) = a;
  *reinterpret_cast<float4*>(l + 4) = b;
#endif
}

__device__ __forceinline__ void stage_fence() {
#if HAVE_ASYNC_LDS
  async_wait0();
#endif
}

// ---------------------------------------------------------------------------
// NT GEMM: C = alpha * A(MxK,row) * B(NxK,row)^T. Block tile 128x128, BK=16.
// 8 waves/WG; each wave owns a 32(M) x 64(N) strip => 8 f32 WMMA accumulators,
// 6 v2f LDS fragment loads feed 8 v_wmma per k-chunk.
// ---------------------------------------------------------------------------
template <int MODE>
__global__ __launch_bounds__(256) void gemm_nt_kernel(
    const float* __restrict__ Abase, const float* __restrict__ Bbase,
    float* __restrict__ Cbase, int K, int lda, int ldb, int ldc, float alpha) {
  __shared__ float sA[BM * BK];  // [m][k]  8 KB
  __shared__ float sB[BN * BK];  // [n][k]  8 KB

  const long z = blockIdx.z;
  const float* A;
  const float* B;
  float* C;
  if (MODE == MODE_PLAIN) {
    A = Abase; B = Bbase; C = Cbase;
  } else if (MODE == MODE_SCORES) {
    const long b = z >> 4, h = z & 15;
    A = Abase + b * (long)L_SEQ * D_MODEL + h * HEAD_DIM;
    B = Bbase + b * (long)L_SEQ * D_MODEL + h * HEAD_DIM;
    C = Cbase + (z << 20);
  } else if (MODE == MODE_ATILDE) {
    const long b = z >> 3, h = z & 7;
    A = Abase + ((b * NUM_HEADS + h) << 20);  // scores JAT slice
    B = A;
    C = Cbase + (z << 20);
  } else {  // MODE_PGEMM: P = S * A_hat^T
    const long b = z >> 3, h = z & 7;
    A = Abase + ((b * NUM_HEADS + h) << 20);  // scores JAT slice
    B = Bbase + (z << 20);                    // A_hat slice
    C = Cbase + (z << 20);
  }

  const int tid = threadIdx.x;
  const int lane = tid & 31;
  const int wave = tid >> 5;
  const int lm = lane & 15;  // fragment row/col within 16
  const int hi = lane >> 4;  // lane half -> K pair / M half
  const int wm = (wave & 3) * 32;   // 4 M-slots of 32
  const int wn = (wave >> 2) * 64;  // 2 N-slots of 64
  const int m0 = blockIdx.y * BM;
  const int n0 = blockIdx.x * BN;

  v8f acc[2][4] = {};

  const int lr = tid >> 1;       // 0..127: tile row staged by this thread
  const int lk = (tid & 1) * 8;  // 0 or 8: K half

  const float* Ap = &A[(long)(m0 + lr) * lda + lk];
  const float* Bp = &B[(long)(n0 + lr) * ldb + lk];
  float* sAp = &sA[lr * BK + lk];
  float* sBp = &sB[lr * BK + lk];

  for (int k0 = 0; k0 < K; k0 += BK) {
    stage8(Ap + k0, sAp);
    stage8(Bp + k0, sBp);
    stage_fence();
    __syncthreads();
#pragma unroll
    for (int kc = 0; kc < 4; ++kc) {
      const int kb = kc * 4 + 2 * hi;  // VGPR i of the frag holds K = kb + i
      const v2f a0 = lds2(&sA[(wm + lm) * BK + kb]);
      const v2f a1 = lds2(&sA[(wm + 16 + lm) * BK + kb]);
#pragma unroll
      for (int j = 0; j < 4; ++j) {
        const v2f bj = lds2(&sB[(wn + 16 * j + lm) * BK + kb]);
        acc[0][j] = wmma4(a0, bj, acc[0][j]);
        acc[1][j] = wmma4(a1, bj, acc[1][j]);
      }
    }
    __syncthreads();
  }

#pragma unroll
  for (int i = 0; i < 2; ++i)
#pragma unroll
    for (int j = 0; j < 4; ++j)
#pragma unroll
      for (int r = 0; r < 8; ++r) {
        const int m = m0 + wm + 16 * i + 8 * hi + r;  // C: vgpr r -> M=r/r+8
        const int n = n0 + wn + 16 * j + lm;
        float v = alpha * acc[i][j][r];
        if (MODE == MODE_ATILDE)  // A_tilde = |U| off-diag, 1 on diag
          v = (m == n) ? 1.0f : fabsf(v);
        C[(long)m * ldc + n] = v;
      }
}

// ---------------------------------------------------------------------------
// Out[b] += coef * sum_{h<8} A_hat[b,h] @ P[b,h]   (NN GEMM, K = 8*1024)
// ---------------------------------------------------------------------------
__global__ __launch_bounds__(256) void jump_accum_kernel(
    const float* __restrict__ Ahat, const float* __restrict__ P,
    float* __restrict__ Out, float coef) {
  __shared__ float sA[BM * BK];  // [m][k]
  __shared__ float sB[BK * BN];  // [k][n]

  const long b = blockIdx.z;
  const float* Ab = Ahat + (b << 23);  // b * 8 * L*L
  const float* Pb = P + (b << 23);
  float* Cb = Out + (b << 20);

  const int tid = threadIdx.x;
  const int lane = tid & 31;
  const int wave = tid >> 5;
  const int lm = lane & 15;
  const int hi = lane >> 4;
  const int wm = (wave & 3) * 32;
  const int wn = (wave >> 2) * 64;
  const int m0 = blockIdx.y * BM;
  const int n0 = blockIdx.x * BN;

  v8f acc[2][4] = {};

  const int lrA = tid >> 1;        // 0..127
  const int lkA = (tid & 1) * 8;   // 0 or 8 (stays inside one head slice)
  const int lrB = tid >> 4;        // 0..15 : K row
  const int lcB = (tid & 15) * 8;  // 0..120: N chunk of 8

  const int Ktot = NUM_JAT * L_SEQ;  // 8192; head slices aligned to BK
  for (int k0 = 0; k0 < Ktot; k0 += BK) {
    {
      const int kk = k0 + lkA, h = kk >> 10, nin = kk & 1023;
      stage8(&Ab[((long)h << 20) + (long)(m0 + lrA) * L_SEQ + nin],
             &sA[lrA * BK + lkA]);
      const int kb = k0 + lrB, hb = kb >> 10, ninb = kb & 1023;
      stage8(&Pb[((long)hb << 20) + (long)ninb * L_SEQ + n0 + lcB],
             &sB[lrB * BN + lcB]);
    }
    stage_fence();
    __syncthreads();
#pragma unroll
    for (int kc = 0; kc < 4; ++kc) {
      const int kb = kc * 4 + 2 * hi;
      const v2f a0 = lds2(&sA[(wm + lm) * BK + kb]);
      const v2f a1 = lds2(&sA[(wm + 16 + lm) * BK + kb]);
#pragma unroll
      for (int j = 0; j < 4; ++j) {
        v2f bj;  // B[k][n]: stride-BN pair -> ds_load_2addr
        bj.x = sB[kb * BN + wn + 16 * j + lm];
        bj.y = sB[(kb + 1) * BN + wn + 16 * j + lm];
        acc[0][j] = wmma4(a0, bj, acc[0][j]);
        acc[1][j] = wmma4(a1, bj, acc[1][j]);
      }
    }
    __syncthreads();
  }

#pragma unroll
  for (int i = 0; i < 2; ++i)
#pragma unroll
    for (int j = 0; j < 4; ++j)
#pragma unroll
      for (int r = 0; r < 8; ++r) {
        const int m = m0 + wm + 16 * i + 8 * hi + r;
        const int n = n0 + wn + 16 * j + lm;
        Cb[(long)m * L_SEQ + n] += coef * acc[i][j][r];
      }
}

// ---------------------------------------------------------------------------
// out[b,l,m] = ( sum_{h>=8} S + 0.5 * sum_{h<8} S ) / 16
// ---------------------------------------------------------------------------
__global__ __launch_bounds__(256) void combine_kernel(
    const float* __restrict__ scores, float* __restrict__ out) {
  const long idx = (long)blockIdx.x * blockDim.x + threadIdx.x;  // < 4*L*L
  const long b = idx >> 20;
  const long i = idx & (LLSZ - 1);
  const float* sb = scores + (b << 24);  // b * 16 * L*L
  float acc = 0.f;
#pragma unroll
  for (int h = 0; h < NUM_HEADS; ++h) {
    const float v = sb[((long)h << 20) + i];
    acc += (h < NUM_JAT) ? 0.5f * v : v;
  }
  out[idx] = acc * (1.0f / 16.0f);
}

// dinv[row] = rsqrt(max(sum_m A_tilde[row,m], 1e-6)); one wave32 per row
__global__ __launch_bounds__(256) void dinv_kernel(const float* __restrict__ A,
                                                   float* __restrict__ dinv) {
  const int lane = threadIdx.x & 31;
  const int w = threadIdx.x >> 5;
  const int row = blockIdx.x * 8 + w;  // < 32*1024
  const float* ar = A + (long)row * L_SEQ;
  float s = 0.f;
  for (int i = lane; i < L_SEQ; i += 32) s += ar[i];
#pragma unroll
  for (int off = 16; off; off >>= 1) s += __shfl_xor(s, off, 32);
  if (lane == 0) dinv[row] = rsqrtf(fmaxf(s, 1e-6f));
}

// A_hat = A_tilde * dinv[l] * dinv[m]   (in place)
__global__ __launch_bounds__(256) void scale_kernel(
    float* __restrict__ A, const float* __restrict__ dinv) {
  const long idx = (long)blockIdx.x * blockDim.x + threadIdx.x;  // < 32*L*L
  const long zl = idx >> 10;  // z*1024 + l
  const long z = idx >> 20;
  const long m = idx & 1023;
  A[idx] *= dinv[zl] * dinv[(z << 10) + m];
}

// ---------------------------------------------------------------------------
extern "C" void kernel_launch(void* const* d_in, const int* in_sizes, int n_in,
                              void* d_out, int out_size, void* d_ws,
                              size_t ws_size, hipStream_t stream) {
  const float* hs = (const float*)d_in[0];
  const float* Wq = (const float*)d_in[1];
  const float* Wk = (const float*)d_in[2];
  float* out = (float*)d_out;
  float* ws = (float*)d_ws;

  const size_t LL = 1ull << 20;
  float* q      = ws;                //  4 * LL
  float* k      = q + 4 * LL;        //  4 * LL
  float* scores = k + 4 * LL;        // 64 * LL  (B,16,L,L)
  float* Abuf   = scores + 64 * LL;  // 32 * LL  (B,8,L,L) A_tilde -> A_hat
  float* Pbuf   = Abuf + 32 * LL;    // 32 * LL  (B,8,L,L) S @ A_hat^T
  float* dinv   = Pbuf + 32 * LL;    // 32 * 1024

  const dim3 blk(256);

  // q = HS @ Wq^T ; k = HS @ Wk^T    (M=4096, N=1024, K=1024)
  gemm_nt_kernel<MODE_PLAIN>
      <<<dim3(D_MODEL / BN, (BATCH * L_SEQ) / BM, 1), blk, 0, stream>>>(
          hs, Wq, q, D_MODEL, D_MODEL, D_MODEL, D_MODEL, 1.0f);
  gemm_nt_kernel<MODE_PLAIN>
      <<<dim3(D_MODEL / BN, (BATCH * L_SEQ) / BM, 1), blk, 0, stream>>>(
          hs, Wk, k, D_MODEL, D_MODEL, D_MODEL, D_MODEL, 1.0f);

  // scores[b,h] = q k^T / sqrt(64)   (K=64, 64 slices)
  gemm_nt_kernel<MODE_SCORES>
      <<<dim3(L_SEQ / BN, L_SEQ / BM, BATCH * NUM_HEADS), blk, 0, stream>>>(
          q, k, scores, HEAD_DIM, D_MODEL, D_MODEL, L_SEQ, 0.125f);

  // out = head-mean of (0.5*S for JAT heads, S for the rest)
  combine_kernel<<<dim3((BATCH << 20) / 256), blk, 0, stream>>>(scores, out);

  // A_tilde = |S S^T / (L*d)| off-diag, 1 on diag   (32 slices, K=1024)
  gemm_nt_kernel<MODE_ATILDE>
      <<<dim3(L_SEQ / BN, L_SEQ / BM, BATCH * NUM_JAT), blk, 0, stream>>>(
          scores, scores, Abuf, L_SEQ, L_SEQ, L_SEQ, L_SEQ,
          1.0f / (1024.0f * 64.0f));

  // degree^-1/2 and symmetric normalization -> A_hat
  dinv_kernel<<<dim3(BATCH * NUM_JAT * L_SEQ / 8), blk, 0, stream>>>(Abuf,
                                                                     dinv);
  scale_kernel<<<dim3((32u << 20) / 256), blk, 0, stream>>>(Abuf, dinv);

  // P = S @ A_hat^T   (32 slices, K=1024)
  gemm_nt_kernel<MODE_PGEMM>
      <<<dim3(L_SEQ / BN, L_SEQ / BM, BATCH * NUM_JAT), blk, 0, stream>>>(
          scores, Abuf, Pbuf, L_SEQ, L_SEQ, L_SEQ, L_SEQ, 1.0f);

  // out += (0.5/16) * sum_h A_hat @ P   (NN, K=8192, one pass per batch)
  jump_accum_kernel<<<dim3(L_SEQ / BN, L_SEQ / BM, BATCH), blk, 0, stream>>>(
      Abuf, Pbuf, out, 0.5f / 16.0f);
}